// SimpleSAGEConv_6124623364516
// MI455X (gfx1250) — compile-verified
//
#include <hip/hip_runtime.h>
#include <math.h>

#define N_NODES_C 50000
#define N_EDGES_C 800000
#define IN_C 32
#define HID 64
#define OUT_C 8
#define N_LAYERS 3

typedef __attribute__((ext_vector_type(2))) float v2f;
typedef __attribute__((ext_vector_type(8))) float v8f;

__device__ __forceinline__ v8f wmma4(v2f a, v2f b, v8f c) {
  // D = A(16x4,f32) x B(4x16,f32) + C(16x16,f32)
  return __builtin_amdgcn_wmma_f32_16x16x4_f32(
      /*neg_a=*/false, a, /*neg_b=*/false, b,
      /*c_mod=*/(short)0, c, /*reuse_a=*/false, /*reuse_b=*/false);
}

__device__ __forceinline__ float gelu_exact(float x) {
  return 0.5f * x * (1.0f + erff(x * 0.70710678118654752440f));
}

// Fragment-ordered weight storage in LDS:
//   element W[k][n] lives at  (((k>>2)*2 + ((k>>1)&1)) * N + n) * 2 + (k&1)
// so that the per-lane B fragment (W[k][n], W[k+1][n]) for k = k0 + 2*hi is
// one contiguous, 8-byte-aligned v2f at (((k0>>2)*2 + hi) * N + n) * 2.
__device__ __forceinline__ int wfrag_store_idx(int k, int n, int N) {
  return (((k >> 2) * 2 + ((k >> 1) & 1)) * N + n) * 2 + (k & 1);
}
__device__ __forceinline__ int wfrag_load_idx(int k0, int hi, int n, int N) {
  return (((k0 >> 2) * 2 + hi) * N + n) * 2;
}

#define ZERO8 {0.f, 0.f, 0.f, 0.f, 0.f, 0.f, 0.f, 0.f}

// ---------------------------------------------------------------- degree
__global__ __launch_bounds__(256) void deg_kernel(const int* __restrict__ ei,
                                                  float* __restrict__ deg,
                                                  int nE) {
  int e = blockIdx.x * blockDim.x + threadIdx.x;
  if (e < nE) {
    int dst = ei[nE + e];
    atomicAdd(&deg[dst], 1.0f);
  }
}

__global__ __launch_bounds__(256) void invdeg_kernel(float* __restrict__ deg,
                                                     int n) {
  int i = blockIdx.x * blockDim.x + threadIdx.x;
  if (i < n) deg[i] = 1.0f / fmaxf(deg[i], 1.0f);
}

// ---------------------------------------------------------------- scatter
// one wave per edge; lane covers channels lane and lane+32
__global__ __launch_bounds__(256) void scatter_kernel(
    const int* __restrict__ ei, const float* __restrict__ h,
    float* __restrict__ agg, int nE) {
  int t = blockIdx.x * blockDim.x + threadIdx.x;
  int e = t >> 5;
  int lane = t & 31;
  if (e < nE) {
    int s = ei[e];
    int d = ei[nE + e];
    float v0 = h[(size_t)s * HID + lane];
    float v1 = h[(size_t)s * HID + lane + 32];
    atomicAdd(&agg[(size_t)d * HID + lane], v0);
    atomicAdd(&agg[(size_t)d * HID + lane + 32], v1);
  }
}

// ---------------------------------------------------------------- encoder
// h = gelu(x @ W1 + b1) @ W2 + b2     (x: n x 32, h: n x 64)
__global__ __launch_bounds__(256) void encoder_kernel(
    const float* __restrict__ x, const float* __restrict__ w1,
    const float* __restrict__ b1, const float* __restrict__ w2,
    const float* __restrict__ b2, float* __restrict__ h, int n) {
  __shared__ float W1f[IN_C * HID];    // fragment order, 8 KB
  __shared__ float W2f[HID * HID];     // fragment order, 16 KB
  __shared__ float Ts[128 * 68];       // 34.8 KB
  __shared__ float B1s[HID];
  __shared__ float B2s[HID];

  const int tid = threadIdx.x;
  for (int i = tid; i < IN_C * HID; i += 256)
    W1f[wfrag_store_idx(i >> 6, i & 63, HID)] = w1[i];
  for (int i = tid; i < HID * HID; i += 256)
    W2f[wfrag_store_idx(i >> 6, i & 63, HID)] = w2[i];
  if (tid < HID) {
    B1s[tid] = b1[tid];
    B2s[tid] = b2[tid];
  }
  __syncthreads();

  const int wave = tid >> 5;
  const int lane = tid & 31;
  const int m = lane & 15;
  const int hi = lane >> 4;
  const int base = blockIdx.x * 128 + wave * 16;
  int rowA = base + m;
  if (rowA >= n) rowA = n - 1;

  // stage 1: T = gelu(x @ W1 + b1), 16x64 tile per wave, K = 32
  v8f acc[4] = {ZERO8, ZERO8, ZERO8, ZERO8};
  const float* xrow = x + (size_t)rowA * IN_C;
  for (int k0 = 0; k0 < IN_C; k0 += 4) {
    v2f a = *(const v2f*)(xrow + k0 + 2 * hi);
#pragma unroll
    for (int t = 0; t < 4; ++t) {
      v2f b = *(const v2f*)&W1f[wfrag_load_idx(k0, hi, t * 16 + m, HID)];
      acc[t] = wmma4(a, b, acc[t]);
    }
  }
#pragma unroll
  for (int t = 0; t < 4; ++t) {
#pragma unroll
    for (int r = 0; r < 8; ++r) {
      const int mo = r + 8 * hi;
      const int no = t * 16 + m;
      Ts[(wave * 16 + mo) * 68 + no] = gelu_exact(acc[t][r] + B1s[no]);
    }
  }
  __syncthreads();

  // stage 2: h = T @ W2 + b2, K = 64
  v8f acc2[4] = {ZERO8, ZERO8, ZERO8, ZERO8};
  const float* trow = &Ts[(wave * 16 + m) * 68];
  for (int k0 = 0; k0 < HID; k0 += 4) {
    v2f a = *(const v2f*)(trow + k0 + 2 * hi);
#pragma unroll
    for (int t = 0; t < 4; ++t) {
      v2f b = *(const v2f*)&W2f[wfrag_load_idx(k0, hi, t * 16 + m, HID)];
      acc2[t] = wmma4(a, b, acc2[t]);
    }
  }
#pragma unroll
  for (int t = 0; t < 4; ++t) {
#pragma unroll
    for (int r = 0; r < 8; ++r) {
      const int mo = r + 8 * hi;
      const int grow = base + mo;
      const int no = t * 16 + m;
      if (grow < n) h[(size_t)grow * HID + no] = acc2[t][r] + B2s[no];
    }
  }
}

// ---------------------------------------------------------------- SAGE layer
// hout = gelu((agg * inv_deg) @ Wl + hin @ Wr + b)
__global__ __launch_bounds__(256) void sage_kernel(
    const float* __restrict__ agg, const float* __restrict__ hin,
    const float* __restrict__ inv_deg, const float* __restrict__ wl,
    const float* __restrict__ wr, const float* __restrict__ bias,
    float* __restrict__ hout, int n) {
  __shared__ float Wlf[HID * HID];  // fragment order, 16 KB
  __shared__ float Wrf[HID * HID];  // fragment order, 16 KB
  __shared__ float Bs[HID];

  const int tid = threadIdx.x;
  for (int i = tid; i < HID * HID; i += 256) {
    const int idx = wfrag_store_idx(i >> 6, i & 63, HID);
    Wlf[idx] = wl[i];
    Wrf[idx] = wr[i];
  }
  if (tid < HID) Bs[tid] = bias[tid];
  __syncthreads();

  const int wave = tid >> 5;
  const int lane = tid & 31;
  const int m = lane & 15;
  const int hi = lane >> 4;
  const int base = blockIdx.x * 128 + wave * 16;
  int rowA = base + m;
  if (rowA >= n) rowA = n - 1;

  const float inv = inv_deg[rowA];
  const float* arow = agg + (size_t)rowA * HID;
  const float* hrow = hin + (size_t)rowA * HID;

  v8f acc[4] = {ZERO8, ZERO8, ZERO8, ZERO8};
  // (agg * inv_deg) @ Wl
  for (int k0 = 0; k0 < HID; k0 += 4) {
    v2f a = *(const v2f*)(arow + k0 + 2 * hi);
    a.x *= inv;
    a.y *= inv;
#pragma unroll
    for (int t = 0; t < 4; ++t) {
      v2f b = *(const v2f*)&Wlf[wfrag_load_idx(k0, hi, t * 16 + m, HID)];
      acc[t] = wmma4(a, b, acc[t]);
    }
  }
  // + hin @ Wr
  for (int k0 = 0; k0 < HID; k0 += 4) {
    v2f a = *(const v2f*)(hrow + k0 + 2 * hi);
#pragma unroll
    for (int t = 0; t < 4; ++t) {
      v2f b = *(const v2f*)&Wrf[wfrag_load_idx(k0, hi, t * 16 + m, HID)];
      acc[t] = wmma4(a, b, acc[t]);
    }
  }
#pragma unroll
  for (int t = 0; t < 4; ++t) {
#pragma unroll
    for (int r = 0; r < 8; ++r) {
      const int mo = r + 8 * hi;
      const int grow = base + mo;
      const int no = t * 16 + m;
      if (grow < n)
        hout[(size_t)grow * HID + no] = gelu_exact(acc[t][r] + Bs[no]);
    }
  }
}

// ---------------------------------------------------------------- decoder
// out = gelu(h @ W1 + b1) @ W2 + b2   (W2: 64x8, zero-padded to 64x16)
__global__ __launch_bounds__(256) void decoder_kernel(
    const float* __restrict__ hin, const float* __restrict__ w1,
    const float* __restrict__ b1, const float* __restrict__ w2,
    const float* __restrict__ b2, float* __restrict__ out, int n) {
  __shared__ float W1f[HID * HID];  // fragment order, 16 KB
  __shared__ float W2f[HID * 16];   // fragment order (N=16), 4 KB
  __shared__ float Ts[128 * 68];    // 34.8 KB
  __shared__ float B1s[HID];
  __shared__ float B2s[16];

  const int tid = threadIdx.x;
  for (int i = tid; i < HID * HID; i += 256)
    W1f[wfrag_store_idx(i >> 6, i & 63, HID)] = w1[i];
  for (int i = tid; i < HID * 16; i += 256) {
    const int r = i >> 4, c = i & 15;
    W2f[wfrag_store_idx(r, c, 16)] = (c < OUT_C) ? w2[r * OUT_C + c] : 0.0f;
  }
  if (tid < 16) B2s[tid] = (tid < OUT_C) ? b2[tid] : 0.0f;
  if (tid < HID) B1s[tid] = b1[tid];
  __syncthreads();

  const int wave = tid >> 5;
  const int lane = tid & 31;
  const int m = lane & 15;
  const int hi = lane >> 4;
  const int base = blockIdx.x * 128 + wave * 16;
  int rowA = base + m;
  if (rowA >= n) rowA = n - 1;

  // stage 1: T = gelu(h @ W1 + b1)
  v8f acc[4] = {ZERO8, ZERO8, ZERO8, ZERO8};
  const float* hrow = hin + (size_t)rowA * HID;
  for (int k0 = 0; k0 < HID; k0 += 4) {
    v2f a = *(const v2f*)(hrow + k0 + 2 * hi);
#pragma unroll
    for (int t = 0; t < 4; ++t) {
      v2f b = *(const v2f*)&W1f[wfrag_load_idx(k0, hi, t * 16 + m, HID)];
      acc[t] = wmma4(a, b, acc[t]);
    }
  }
#pragma unroll
  for (int t = 0; t < 4; ++t) {
#pragma unroll
    for (int r = 0; r < 8; ++r) {
      const int mo = r + 8 * hi;
      const int no = t * 16 + m;
      Ts[(wave * 16 + mo) * 68 + no] = gelu_exact(acc[t][r] + B1s[no]);
    }
  }
  __syncthreads();

  // stage 2: out = T @ W2 + b2, single 16x16 N-tile (cols 8..15 zero pad)
  v8f acc2 = ZERO8;
  const float* trow = &Ts[(wave * 16 + m) * 68];
  for (int k0 = 0; k0 < HID; k0 += 4) {
    v2f a = *(const v2f*)(trow + k0 + 2 * hi);
    v2f b = *(const v2f*)&W2f[wfrag_load_idx(k0, hi, m, 16)];
    acc2 = wmma4(a, b, acc2);
  }
#pragma unroll
  for (int r = 0; r < 8; ++r) {
    const int mo = r + 8 * hi;
    const int grow = base + mo;
    if (grow < n && m < OUT_C)
      out[(size_t)grow * OUT_C + m] = acc2[r] + B2s[m];
  }
}

// ---------------------------------------------------------------- launch
extern "C" void kernel_launch(void* const* d_in, const int* in_sizes, int n_in,
                              void* d_out, int out_size, void* d_ws,
                              size_t ws_size, hipStream_t stream) {
  const float* x = (const float*)d_in[0];
  const int* ei = (const int*)d_in[1];  // (2, E) int32 row-major
  const float* enc_w1 = (const float*)d_in[2];
  const float* enc_b1 = (const float*)d_in[3];
  const float* enc_w2 = (const float*)d_in[4];
  const float* enc_b2 = (const float*)d_in[5];
  const float* sage_wl = (const float*)d_in[6];
  const float* sage_wr = (const float*)d_in[7];
  const float* sage_b = (const float*)d_in[8];
  const float* dec_w1 = (const float*)d_in[9];
  const float* dec_b1 = (const float*)d_in[10];
  const float* dec_w2 = (const float*)d_in[11];
  const float* dec_b2 = (const float*)d_in[12];
  float* out = (float*)d_out;

  const int nN = N_NODES_C;
  const int nE = N_EDGES_C;

  // workspace carve: h0, h1, agg (nN*64 each), deg (nN)  ~38.6 MB
  float* h0 = (float*)d_ws;
  float* h1 = h0 + (size_t)nN * HID;
  float* agg = h1 + (size_t)nN * HID;
  float* deg = agg + (size_t)nN * HID;

  // degrees -> inverse degrees (in place)
  hipMemsetAsync(deg, 0, (size_t)nN * sizeof(float), stream);
  deg_kernel<<<(nE + 255) / 256, 256, 0, stream>>>(ei, deg, nE);
  invdeg_kernel<<<(nN + 255) / 256, 256, 0, stream>>>(deg, nN);

  const int nb = (nN + 127) / 128;
  encoder_kernel<<<nb, 256, 0, stream>>>(x, enc_w1, enc_b1, enc_w2, enc_b2, h0,
                                         nN);

  float* cur = h0;
  float* nxt = h1;
  for (int l = 0; l < N_LAYERS; ++l) {
    hipMemsetAsync(agg, 0, (size_t)nN * HID * sizeof(float), stream);
    scatter_kernel<<<(nE * 32 + 255) / 256, 256, 0, stream>>>(ei, cur, agg, nE);
    sage_kernel<<<nb, 256, 0, stream>>>(agg, cur, deg, sage_wl + l * HID * HID,
                                        sage_wr + l * HID * HID,
                                        sage_b + l * HID, nxt, nN);
    float* t = cur;
    cur = nxt;
    nxt = t;
  }

  decoder_kernel<<<nb, 256, 0, stream>>>(cur, dec_w1, dec_b1, dec_w2, dec_b2,
                                         out, nN);
}